// RecurrentNN_19439021981769
// MI455X (gfx1250) — compile-verified
//
#include <hip/hip_runtime.h>

typedef __attribute__((ext_vector_type(2))) float v2f;
typedef __attribute__((ext_vector_type(8))) float v8f;

#define T_LEN 4096
#define B_LEN 2048
#define HID   10
#define OUTD  5

// Hardware tanh (V_TANH_F32, gfx1250 "tanh-insts"); fall back to libm if absent.
#if __has_builtin(__builtin_amdgcn_tanhf)
  #define HW_TANH(x) __builtin_amdgcn_tanhf(x)
#elif __has_builtin(__builtin_amdgcn_tanh_f32)
  #define HW_TANH(x) __builtin_amdgcn_tanh_f32(x)
#else
  #define HW_TANH(x) tanhf(x)
#endif

// ---------------- Phase 0: gather x[:, B-1, 0] into contiguous xc ----------------
__global__ void gather_x_kernel(const float* __restrict__ x, float* __restrict__ xc) {
    int t = blockIdx.x * blockDim.x + threadIdx.x;
    if (t < T_LEN) xc[t] = x[(size_t)t * B_LEN + (B_LEN - 1)];
}

// ---------------- Phase 1: single-wave serial scan using V_WMMA_F32_16X16X4_F32 ----
// Layout trick: A[m][k] = h[k] (rows replicated)  -> D[m][n] = sum_k h[k]*W_hh[n][k]
// replicated over rows, so lane l owns hidden unit n = l%16 in every D VGPR.
// The 3 K-chunks use INDEPENDENT accumulators (C=0) so the WMMAs pipeline instead
// of serializing through the C operand; element-0 results are tree-summed in VALU.
__global__ void rnn_scan_kernel(const float* __restrict__ xc,
                                const float* __restrict__ W_ih,
                                const float* __restrict__ W_hh,
                                const float* __restrict__ b_ih,
                                const float* __restrict__ b_hh,
                                float* __restrict__ Hws) {
    const int lane = threadIdx.x;      // 0..31, one wave
    const int n    = lane & 15;        // hidden unit owned by this lane
    const int half = lane >> 4;

    const float wih_n  = (n < HID) ? W_ih[n] : 0.0f;
    const float bias_n = (n < HID) ? (b_ih[n] + b_hh[n]) : 0.0f;

    // B chunks (2 VGPRs each): B[k][n] = W_hh[n][k], zero padded.
    // f32 B layout mirrors A: lane -> N, (vgpr, lane-half) -> K.
    v2f Bm[3];
    int idxb[3][2];   // ds_bpermute byte indices to build A[m][k] = h[k]
    #pragma unroll
    for (int c = 0; c < 3; ++c) {
        #pragma unroll
        for (int v = 0; v < 2; ++v) {
            int K = 4 * c + 2 * half + v;
            Bm[c][v]   = (K < HID && n < HID) ? W_hh[n * HID + K] : 0.0f;
            idxb[c][v] = K * 4;   // h[K] lives in lane K (bpermute index in bytes)
        }
    }

    float hcur = 0.0f;   // h0 = 0; lanes n>=10 stay exactly 0 forever

    for (int t = 0; t < T_LEN; ++t) {
        // prefetch future x (contiguous stream) into near caches
        __builtin_prefetch(xc + t + 128, 0, 3);
        float xt = xc[t];

        // Build A chunks from hcur via lane-permute (no LDS memory used)
        int hb = __float_as_int(hcur);
        v2f A0, A1, A2;
        A0.x = __int_as_float(__builtin_amdgcn_ds_bpermute(idxb[0][0], hb));
        A0.y = __int_as_float(__builtin_amdgcn_ds_bpermute(idxb[0][1], hb));
        A1.x = __int_as_float(__builtin_amdgcn_ds_bpermute(idxb[1][0], hb));
        A1.y = __int_as_float(__builtin_amdgcn_ds_bpermute(idxb[1][1], hb));
        A2.x = __int_as_float(__builtin_amdgcn_ds_bpermute(idxb[2][0], hb));
        A2.y = __int_as_float(__builtin_amdgcn_ds_bpermute(idxb[2][1], hb));

        // Three independent WMMAs (no C-chain) -> pipelined, single WMMA latency
        v8f z = {};
        v8f acc0 = __builtin_amdgcn_wmma_f32_16x16x4_f32(false, A0, false, Bm[0],
                                                         (short)0, z, false, false);
        v8f acc1 = __builtin_amdgcn_wmma_f32_16x16x4_f32(false, A1, false, Bm[1],
                                                         (short)0, z, false, false);
        v8f acc2 = __builtin_amdgcn_wmma_f32_16x16x4_f32(false, A2, false, Bm[2],
                                                         (short)0, z, false, false);

        // rows replicated: element 0 of any D holds column n for this lane
        float pre = (acc0[0] + acc1[0]) + (acc2[0] + fmaf(xt, wih_n, bias_n));
        float h   = HW_TANH(pre);         // single V_TANH_F32; lanes n>=10 stay 0
        hcur = h;

        // lanes l and l+16 write identical value to identical address (benign)
        Hws[t * 16 + n] = h;
    }
}

// ---------------- Phase 2: out = H(4096x10) @ W_out^T + b_out via WMMA tiles -------
__global__ void out_gemm_kernel(const float* __restrict__ Hws,
                                const float* __restrict__ W_out,
                                const float* __restrict__ b_out,
                                float* __restrict__ out) {
    const int lane = threadIdx.x & 31;
    const int wave = threadIdx.x >> 5;
    const int tile = blockIdx.x * (blockDim.x >> 5) + wave;   // 256 tiles total
    const int n    = lane & 15;
    const int half = lane >> 4;
    const int m0   = tile * 16;

    // B[k][n] = W_out[n][k], zero padded
    v2f Bm[3];
    #pragma unroll
    for (int c = 0; c < 3; ++c) {
        #pragma unroll
        for (int v = 0; v < 2; ++v) {
            int K = 4 * c + 2 * half + v;
            Bm[c][v] = (K < HID && n < OUTD) ? W_out[n * HID + K] : 0.0f;
        }
    }

    // C seeded with bias broadcast down the rows
    float cb = (n < OUTD) ? b_out[n] : 0.0f;
    v8f acc;
    #pragma unroll
    for (int i = 0; i < 8; ++i) acc[i] = cb;

    // A tile: lane -> row m0 + n (M = lane%16), (vgpr, half) -> K ; H row stride 16
    const float* row = Hws + (size_t)(m0 + n) * 16;
    #pragma unroll
    for (int c = 0; c < 3; ++c) {
        v2f a = *(const v2f*)(row + 4 * c + 2 * half);   // 8B aligned
        acc = __builtin_amdgcn_wmma_f32_16x16x4_f32(false, a, false, Bm[c],
                                                    (short)0, acc, false, false);
    }

    // D: vgpr r holds row (m0 + r + 8*half), column n
    if (n < OUTD) {
        #pragma unroll
        for (int r = 0; r < 8; ++r) {
            out[(size_t)(m0 + r + 8 * half) * OUTD + n] = acc[r];
        }
    }
}

extern "C" void kernel_launch(void* const* d_in, const int* in_sizes, int n_in,
                              void* d_out, int out_size, void* d_ws, size_t ws_size,
                              hipStream_t stream) {
    const float* x     = (const float*)d_in[0];
    const float* W_ih  = (const float*)d_in[1];
    const float* W_hh  = (const float*)d_in[2];
    const float* b_ih  = (const float*)d_in[3];
    const float* b_hh  = (const float*)d_in[4];
    const float* W_out = (const float*)d_in[5];
    const float* b_out = (const float*)d_in[6];
    float* out = (float*)d_out;

    float* xc  = (float*)d_ws;        // T_LEN floats          (16 KB)
    float* Hws = xc + T_LEN;          // T_LEN*16 floats       (256 KB)

    gather_x_kernel<<<T_LEN / 256, 256, 0, stream>>>(x, xc);
    rnn_scan_kernel<<<1, 32, 0, stream>>>(xc, W_ih, W_hh, b_ih, b_hh, Hws);
    out_gemm_kernel<<<32, 256, 0, stream>>>(Hws, W_out, b_out, out);
}